// SecondaryCapsule_12292196401267
// MI455X (gfx1250) — compile-verified
//
#include <hip/hip_runtime.h>
#include <math.h>

// CapsNet dynamic routing, fused 3-pass formulation for gfx1250 (MI455X).
//   pass 0 (uniform):  v1 = squash(mean_p u_hat)
//   pass 1 (weighted): logit = u.v1      ; v2 = squash(sum exp(logit) u / Z)
//   pass 2 (weighted): logit = u.(v1+v2) ; v3 = squash(sum exp(logit) u / Z)
// u_hat is recomputed per pass via V_WMMA_F32_16X16X4_F32 (M = 16-batch tile,
// N = Ds = 16, K = Dp = 8 -> two chained wmma ops). W (18.9 MB) and x (4.7 MB)
// stay L2-resident; the 302 MB u_hat tensor never touches HBM.

typedef float v2f __attribute__((ext_vector_type(2)));
typedef float v8f __attribute__((ext_vector_type(8)));

constexpr int BTOT = 128;   // batch
constexpr int PC   = 1152;  // primary caps
constexpr int DP   = 8;     // primary dim
constexpr int SC   = 32;    // secondary caps
constexpr int DSC  = 16;    // secondary dim
constexpr int BT   = 16;    // batch tile (= WMMA M)
constexpr int THREADS = 1024;
constexpr int WAVES   = THREADS / 32;   // 32 waves
constexpr int PPW     = PC / WAVES;     // 36 p per wave per pass
constexpr float EPS   = 1e-7f;

// xor-swap within the 32-lane group via ds_swizzle immediate patterns
// (offset = xor_mask<<10 | and_mask 0x1F); masks < 16 stay within each half.
#define SWZ_XOR_ADD(t, imm) \
    (t) += __int_as_float(__builtin_amdgcn_ds_swizzle(__float_as_int(t), (imm)))

__device__ __forceinline__ float hsum16(float v) {
    SWZ_XOR_ADD(v, 0x041F);   // xor 1
    SWZ_XOR_ADD(v, 0x081F);   // xor 2
    SWZ_XOR_ADD(v, 0x101F);   // xor 4
    SWZ_XOR_ADD(v, 0x201F);   // xor 8
    return v;
}

// Fragment loads for one p. A f32 16x4 layout: v2f {K=2h, K=2h+1} per lane,
// second WMMA +4 in K; B symmetric with B[k][n] = W[s,p,n,k].
#define LOAD_FRAGS(xp, wp, a0, a1, b0, b1)            \
    do {                                              \
        const v2f* ax_ = (const v2f*)(xp);            \
        const v2f* bw_ = (const v2f*)(wp);            \
        a0 = ax_[half];  a1 = ax_[2 + half];          \
        b0 = bw_[half];  b1 = bw_[2 + half];          \
    } while (0)

// One weighted routing pass: recompute u_hat, logit[b][p] = u_hat[b,p,:].vreg,
// accumulate S += exp(logit)*u_hat and Z += exp(logit); reduce across waves.
__device__ __forceinline__ void weighted_pass(
    const float* __restrict__ xrow, const float* __restrict__ wrow,
    int pBeg, int pEnd, int half, int l16, int tid,
    const float vreg[8], float (*lds_S)[DSC], float* lds_Z)
{
    if (tid < BT * DSC) {
        lds_S[tid >> 4][tid & 15] = 0.0f;
        if (tid < BT) lds_Z[tid] = 0.0f;
    }
    __syncthreads();

    v8f Sacc = {};
    float zacc[8];
    #pragma unroll
    for (int r = 0; r < 8; ++r) zacc[r] = 0.0f;

    for (int p = pBeg; p < pEnd; ++p) {
        v2f a0, a1, b0, b1;
        LOAD_FRAGS(xrow + (size_t)p * DP, wrow + (size_t)p * DSC * DP, a0, a1, b0, b1);
        v8f D = {};
        D = __builtin_amdgcn_wmma_f32_16x16x4_f32(false, a0, false, b0, (short)0, D, false, false);
        D = __builtin_amdgcn_wmma_f32_16x16x4_f32(false, a1, false, b1, (short)0, D, false, false);

        // 8 independent j-reductions, stage-batched so 8 ds_swizzles are in
        // flight per stage (hides DS latency; no per-row branches).
        float t[8];
        #pragma unroll
        for (int r = 0; r < 8; ++r) t[r] = D[r] * vreg[r];
        #pragma unroll
        for (int r = 0; r < 8; ++r) SWZ_XOR_ADD(t[r], 0x041F);
        #pragma unroll
        for (int r = 0; r < 8; ++r) SWZ_XOR_ADD(t[r], 0x081F);
        #pragma unroll
        for (int r = 0; r < 8; ++r) SWZ_XOR_ADD(t[r], 0x101F);
        #pragma unroll
        for (int r = 0; r < 8; ++r) SWZ_XOR_ADD(t[r], 0x201F);
        #pragma unroll
        for (int r = 0; r < 8; ++r) {
            float wgt = __expf(t[r]);    // logits ~1e-4: max-free softmax safe
            Sacc[r] += wgt * D[r];
            zacc[r] += wgt;
        }
    }

    #pragma unroll
    for (int r = 0; r < 8; ++r)
        atomicAdd(&lds_S[r + half * 8][l16], Sacc[r]);
    if (l16 == 0) {
        #pragma unroll
        for (int r = 0; r < 8; ++r) atomicAdd(&lds_Z[r + half * 8], zacc[r]);
    }
    __syncthreads();
}

__global__ __launch_bounds__(THREADS)
void caps_route_kernel(const float* __restrict__ x,   // (B, P, Dp)
                       const float* __restrict__ W,   // (S, P, Ds, Dp)
                       float* __restrict__ out)       // (B, S, Ds)
{
    __shared__ float lds_S[BT][DSC];
    __shared__ float lds_Z[BT];
    __shared__ float lds_v[2][BT][DSC];   // v1, v2

    const int s    = blockIdx.x;
    const int b0   = blockIdx.y * BT;
    const int tid  = threadIdx.x;
    const int wave = tid >> 5;
    const int lane = tid & 31;
    const int half = lane >> 4;
    const int l16  = lane & 15;

    const float* xrow = x + (size_t)(b0 + l16) * PC * DP;                 // A: M=batch
    const float* wrow = W + (size_t)s * PC * DSC * DP + (size_t)l16 * DP; // B[k][n]=W[s,p,n,k]
    const int pBeg = wave * PPW, pEnd = pBeg + PPW;

    //===== pass 0: uniform coupling; chain the p-sum straight through the
    //===== WMMA accumulator (no separate vector adds); Z == PC exactly.
    if (tid < BT * DSC) lds_S[tid >> 4][tid & 15] = 0.0f;
    __syncthreads();
    {
        v8f Sacc = {};
        for (int p = pBeg; p < pEnd; ++p) {
            v2f a0, a1, b0, b1;
            LOAD_FRAGS(xrow + (size_t)p * DP, wrow + (size_t)p * DSC * DP, a0, a1, b0, b1);
            Sacc = __builtin_amdgcn_wmma_f32_16x16x4_f32(false, a0, false, b0, (short)0, Sacc, false, false);
            Sacc = __builtin_amdgcn_wmma_f32_16x16x4_f32(false, a1, false, b1, (short)0, Sacc, false, false);
        }
        #pragma unroll
        for (int r = 0; r < 8; ++r)
            atomicAdd(&lds_S[r + half * 8][l16], Sacc[r]);
    }
    __syncthreads();
    if (tid < BT * DSC) {
        const int b = tid >> 4, j = tid & 15;
        float sv  = lds_S[b][j] * (1.0f / (float)PC);
        float nsq = hsum16(sv * sv);
        float nrm = sqrtf(nsq);
        lds_v[0][b][j] = (nsq / (1.0f + nsq)) * sv / (nrm + EPS);
    }
    __syncthreads();

    // per-lane v in WMMA D layout: row r <-> batch r + half*8, column j = l16
    float vreg[8];
    #pragma unroll
    for (int r = 0; r < 8; ++r) vreg[r] = lds_v[0][r + half * 8][l16];

    //===== pass 1: logits = u.v1 =====
    weighted_pass(xrow, wrow, pBeg, pEnd, half, l16, tid, vreg, lds_S, lds_Z);
    if (tid < BT * DSC) {
        const int b = tid >> 4, j = tid & 15;
        float sv  = lds_S[b][j] / lds_Z[b];
        float nsq = hsum16(sv * sv);
        float nrm = sqrtf(nsq);
        lds_v[1][b][j] = (nsq / (1.0f + nsq)) * sv / (nrm + EPS);
    }
    __syncthreads();

    //===== pass 2: logits = u.v1 + u.v2 = u.(v1+v2) -> single reduction =====
    #pragma unroll
    for (int r = 0; r < 8; ++r) vreg[r] += lds_v[1][r + half * 8][l16];
    weighted_pass(xrow, wrow, pBeg, pEnd, half, l16, tid, vreg, lds_S, lds_Z);
    if (tid < BT * DSC) {
        const int b = tid >> 4, j = tid & 15;
        float sv  = lds_S[b][j] / lds_Z[b];
        float nsq = hsum16(sv * sv);
        float nrm = sqrtf(nsq);
        float vv  = (nsq / (1.0f + nsq)) * sv / (nrm + EPS);
        out[(((size_t)(b0 + b)) * SC + s) * DSC + j] = vv;
    }
}

extern "C" void kernel_launch(void* const* d_in, const int* in_sizes, int n_in,
                              void* d_out, int out_size, void* d_ws, size_t ws_size,
                              hipStream_t stream) {
    const float* x = (const float*)d_in[0];   // (128, 1152, 8) f32
    const float* W = (const float*)d_in[1];   // (32, 1152, 16, 8) f32
    float* out = (float*)d_out;               // (128, 32, 16) f32

    dim3 grid(SC, BTOT / BT);                 // 32 x 8 = 256 workgroups
    caps_route_kernel<<<grid, THREADS, 0, stream>>>(x, W, out);
}